// ODLinear_36867999268891
// MI455X (gfx1250) — compile-verified
//
#include <hip/hip_runtime.h>
#include <stdint.h>

typedef float v2f  __attribute__((ext_vector_type(2)));
typedef float v8f  __attribute__((ext_vector_type(8)));
typedef unsigned int u32x4 __attribute__((ext_vector_type(4)));
typedef int   i32x4 __attribute__((ext_vector_type(4)));
typedef int   i32x8 __attribute__((ext_vector_type(8)));

#if defined(__has_builtin)
#if __has_builtin(__builtin_amdgcn_tensor_load_to_lds)
#define HAVE_TDM 1
#else
#define HAVE_TDM 0
#endif
#else
#define HAVE_TDM 0
#endif

#define T_IO    48          // T_IN == T_OUT == 48
#define NCH     4096        // channels (64*64)
#define CH      4           // channels per workgroup (one per wave)
#define BT      16          // batch tile (WMMA M)
#define THREADS 128         // 4 waves of 32

#if HAVE_TDM
// Build a 2D TDM descriptor (D#) and issue tensor_load_to_lds (6-arg form:
// group0, group1, group2, group3, extra group, cpol).
// tile_d0 elements contiguous per row, tile_d1 rows, row stride in elements.
__device__ __forceinline__ void tdm_load_2d(uint32_t lds_off, const void* gptr,
                                            uint32_t tile_d0, uint32_t tile_d1,
                                            uint64_t stride_d0_elems) {
  const uint64_t ga = (uint64_t)(uintptr_t)gptr;
  const uint32_t td0 = 1u << 30, td1 = 1u << 30;   // huge tensor dims: no OOB clipping

  u32x4 g0;
  g0[0] = 1u;                                           // count=1, user mode
  g0[1] = lds_off;                                      // lds_addr (bytes)
  g0[2] = (uint32_t)ga;                                 // global_addr[31:0]
  g0[3] = (uint32_t)((ga >> 32) & 0x1FFFFFFu)           // global_addr[56:32]
        | (2u << 30);                                   // type = 2 ("image")

  i32x8 g1;
  g1[0] = (int)(2u << 16);                              // wg_mask=0, data_size=2 (4B)
  g1[1] = (int)((td0 & 0xFFFFu) << 16);                 // tensor_dim0[15:0]
  g1[2] = (int)((td0 >> 16) | ((td1 & 0xFFFFu) << 16)); // td0[31:16] | td1[15:0]
  g1[3] = (int)((td1 >> 16) | (tile_d0 << 16));         // td1[31:16] | tile_dim0
  g1[4] = (int)(tile_d1 & 0xFFFFu);                     // tile_dim1, tile_dim2=0
  g1[5] = (int)(uint32_t)(stride_d0_elems & 0xFFFFFFFFu);        // dim0 stride lo
  g1[6] = (int)(uint32_t)((stride_d0_elems >> 32) & 0xFFFFu);    // dim0 stride hi
  g1[7] = 0;

  i32x4 z4 = {0, 0, 0, 0};
  i32x8 z8 = {0, 0, 0, 0, 0, 0, 0, 0};
  __builtin_amdgcn_tensor_load_to_lds(g0, g1, z4, z4, z8, 0);
}
#endif

__global__ __launch_bounds__(THREADS)
void odlinear_wmma_f32(const float* __restrict__ x,   // [B*T, 4096]
                       const float* __restrict__ W,   // [4096, 48, 48] (o major, t minor)
                       const float* __restrict__ bias,// [4096, 48]
                       float* __restrict__ out) {     // [B*T, 4096]
  __shared__ float Wbuf[CH * T_IO * T_IO];   // per-wave channel: [o][t]   36 KB
  __shared__ float Xbuf[BT * T_IO * CH];     // [b_local][t][ch]           12 KB (reused for out)

  const int tid  = (int)threadIdx.x;
  const int wv   = __builtin_amdgcn_readfirstlane(tid >> 5); // wave id 0..3 (SGPR)
  const int lane = tid & 31;
  const int n    = lane & 15;       // WMMA M row (A) == N col (B/D)
  const int half = lane >> 4;       // K-pair selector for 16x16x4 f32 fragments
  const int cbase = (int)blockIdx.x * CH;
  const int bbase = (int)blockIdx.y * BT;
  const int c     = cbase + wv;     // this wave's channel

  // ---------------- stage W + x tile into LDS ----------------
#if HAVE_TDM
  // W channel block: 2304 contiguous floats -> Wbuf[wv]
  tdm_load_2d((uint32_t)(uintptr_t)&Wbuf[wv * (T_IO * T_IO)],
              W + (size_t)c * (T_IO * T_IO),
              (uint32_t)(T_IO * T_IO), 1u, (uint64_t)(T_IO * T_IO));
  // x tile: this wave stages 192 of the 768 (b,t) rows, CH=4 floats each,
  // row stride 4096 elements -> Xbuf[row][ch]
  tdm_load_2d((uint32_t)(uintptr_t)&Xbuf[wv * 192 * CH],
              x + ((size_t)(bbase * T_IO + wv * 192)) * NCH + cbase,
              (uint32_t)CH, 192u, (uint64_t)NCH);
  __builtin_amdgcn_s_wait_tensorcnt(0);
#else
  for (int i = tid; i < CH * T_IO * T_IO; i += THREADS)
    Wbuf[i] = W[(size_t)cbase * (T_IO * T_IO) + i];
  for (int i = tid; i < BT * T_IO * CH; i += THREADS) {
    const int row = i >> 2, col = i & 3;
    Xbuf[i] = x[((size_t)(bbase * T_IO + row)) * NCH + cbase + col];
  }
#endif
  __syncthreads();

  // ---------------- accumulators, bias folded into init ----------------
  const float* bc = bias + (size_t)c * T_IO;
  const float bv0 = bc[ 0 + n], bv1 = bc[16 + n], bv2 = bc[32 + n];
  v8f acc0, acc1, acc2;
#pragma unroll
  for (int r = 0; r < 8; ++r) { acc0[r] = bv0; acc1[r] = bv1; acc2[r] = bv2; }

  // ---------------- K loop: 12 steps of v_wmma_f32_16x16x4_f32 ----------------
  const float* wb = &Wbuf[wv * (T_IO * T_IO)];
#pragma unroll
  for (int kk = 0; kk < 12; ++kk) {
    const int t0 = kk * 4 + half * 2;   // lane's K pair: {t0, t0+1}
    v2f a, b0, b1, b2;
    // A fragment: X[b_local = n][t0, t0+1] for this channel
    a.x = Xbuf[(n * T_IO + t0    ) * CH + wv];
    a.y = Xbuf[(n * T_IO + t0 + 1) * CH + wv];
    // B fragments: W[o = jtile*16 + n][t0, t0+1]  (contiguous pair -> ds b64)
    b0.x = wb[( 0 + n) * T_IO + t0];  b0.y = wb[( 0 + n) * T_IO + t0 + 1];
    b1.x = wb[(16 + n) * T_IO + t0];  b1.y = wb[(16 + n) * T_IO + t0 + 1];
    b2.x = wb[(32 + n) * T_IO + t0];  b2.y = wb[(32 + n) * T_IO + t0 + 1];
    acc0 = __builtin_amdgcn_wmma_f32_16x16x4_f32(false, a, false, b0, (short)0, acc0, false, false);
    acc1 = __builtin_amdgcn_wmma_f32_16x16x4_f32(false, a, false, b1, (short)0, acc1, false, false);
    acc2 = __builtin_amdgcn_wmma_f32_16x16x4_f32(false, a, false, b2, (short)0, acc2, false, false);
  }

  // ---------------- transpose through LDS for coalesced stores ----------------
  __syncthreads();   // everyone done reading Xbuf
#pragma unroll
  for (int r = 0; r < 8; ++r) {
    const int bl = r + half * 8;       // D element (VGPR r, lane) -> M = r + 8*half
    Xbuf[(bl * T_IO +  0 + n) * CH + wv] = acc0[r];
    Xbuf[(bl * T_IO + 16 + n) * CH + wv] = acc1[r];
    Xbuf[(bl * T_IO + 32 + n) * CH + wv] = acc2[r];
  }
  __syncthreads();
  for (int i = tid; i < BT * T_IO * CH; i += THREADS) {
    const int row = i >> 2, col = i & 3;
    out[((size_t)(bbase * T_IO + row)) * NCH + cbase + col] = Xbuf[i];
  }
}

extern "C" void kernel_launch(void* const* d_in, const int* in_sizes, int n_in,
                              void* d_out, int out_size, void* d_ws, size_t ws_size,
                              hipStream_t stream) {
  (void)in_sizes; (void)n_in; (void)out_size; (void)d_ws; (void)ws_size;
  // setup_inputs order: x, te, se, pe, adj, s_adj, W, b  (only x, W, b are used)
  const float* x = (const float*)d_in[0];
  const float* W = (const float*)d_in[6];
  const float* b = (const float*)d_in[7];
  float* out = (float*)d_out;

  dim3 grid(NCH / CH, 128 / BT);   // (1024, 8)
  odlinear_wmma_f32<<<grid, THREADS, 0, stream>>>(x, W, b, out);
}